// Multi_Attention_32023276159238
// MI455X (gfx1250) — compile-verified
//
#include <hip/hip_runtime.h>
#include <hip/hip_bf16.h>
#include <math.h>

typedef __attribute__((ext_vector_type(2))) float v2f;
typedef __attribute__((ext_vector_type(4))) float v4f;
typedef __attribute__((ext_vector_type(8))) float v8f;

#define HW       4096   // 64*64
#define C_PER_IN 128
#define CTOT     512
#define BATCH    16
#define NPLANE   (BATCH * CTOT)  // 8192
#define PAD      32

// ---------------------------------------------------------------------------
// Kernel 1: per-(b,c) plane mean of map1 = concat(x1..x4, axis=1).
// One 256-thread block per plane; b128 loads; wave32 shfl reduce.
// Regular-temporal loads on purpose: pull the 134MB input into the 192MB L2
// so kernel 3's re-read hits cache.
// ---------------------------------------------------------------------------
__global__ __launch_bounds__(256) void k_reduce(
    const float* __restrict__ x1, const float* __restrict__ x2,
    const float* __restrict__ x3, const float* __restrict__ x4,
    float* __restrict__ gp)
{
    __shared__ float smem[8];
    const int plane = blockIdx.x;
    const int b = plane >> 9;
    const int c = plane & 511;
    const int sel = c >> 7;
    const float* src = (sel == 0) ? x1 : (sel == 1) ? x2 : (sel == 2) ? x3 : x4;
    src += (size_t)(b * C_PER_IN + (c & 127)) * HW;

    const v4f* p = (const v4f*)src;
    const int t = threadIdx.x;
    float s = 0.f;
#pragma unroll
    for (int k = 0; k < 4; ++k) {
        v4f v = p[t + k * 256];
        s += (v.x + v.y) + (v.z + v.w);
    }
    // wave32 butterfly reduce
#pragma unroll
    for (int off = 16; off >= 1; off >>= 1)
        s += __shfl_xor(s, off, 32);
    if ((t & 31) == 0) smem[t >> 5] = s;
    __syncthreads();
    if (t == 0) {
        float tot = 0.f;
#pragma unroll
        for (int w = 0; w < 8; ++w) tot += smem[w];
        gp[plane] = tot * (1.0f / (float)HW);
    }
}

// ---------------------------------------------------------------------------
// Kernel 2: dilated convs over the channel axis + relu + linear + sigmoid.
// One wave32 block per batch. Lane layout matches the 16x4 f32 WMMA A-matrix:
//   lanes 0-15  : M = lane,      A regs = K0,K1  (branches 0,1)
//   lanes 16-31 : M = lane - 16, A regs = K2,K3  (branches 2,3)
// A[m,k] = relu(conv_k(channel m)) * lin_w[k];  B = ones so every column of
// D holds sum_k A[m,k] regardless of B's lane layout. D extracted per the
// documented C/D layout (VGPR r -> M=r for lanes 0-15, M=r+8 for lanes 16-31).
// ---------------------------------------------------------------------------
__global__ __launch_bounds__(32) void k_gate(
    const float* __restrict__ gp,
    const float* __restrict__ conv_w,  // (4,1,1,9) flat
    const float* __restrict__ conv_b,  // (4,)
    const float* __restrict__ lin_w,   // (1,4) flat
    const float* __restrict__ lin_b,   // (1,)
    float* __restrict__ gate)
{
    __shared__ float sgp[PAD + CTOT + PAD];
    const int b = blockIdx.x;
    const int l = threadIdx.x;  // lane id, wave32

    // zero pads (32 each side) + stage gp row in LDS
    sgp[l] = 0.f;
    sgp[PAD + CTOT + l] = 0.f;
    const float* gpb = gp + b * CTOT;
#pragma unroll
    for (int j = 0; j < 16; ++j)
        sgp[PAD + l + j * 32] = gpb[l + j * 32];
    __syncthreads();

    // this lane's two branches
    const int m  = (l < 16) ? l : (l - 16);
    const int k0 = (l < 16) ? 0 : 2;
    const int k1 = k0 + 1;
    const int d0 = 1 << k0;   // dilation = 1,2,4,8 ; padding = 4*d (centered)
    const int d1 = 1 << k1;

    float w0[9], w1[9];
#pragma unroll
    for (int t = 0; t < 9; ++t) {
        w0[t] = conv_w[k0 * 9 + t];
        w1[t] = conv_w[k1 * 9 + t];
    }
    const float cb0 = conv_b[k0], cb1 = conv_b[k1];
    const float lw0 = lin_w[k0],  lw1 = lin_w[k1];
    const float lb  = lin_b[0];

    v2f bones; bones.x = 1.0f; bones.y = 1.0f;

    // D-matrix extraction mapping: lanes 0-7 write M=lane (elem lane),
    // lanes 16-23 write M=8+(lane-16) (elem lane-16).
    const int  idx      = (l < 8) ? l : (l - 16);
    const bool do_store = (l < 8) || (l >= 16 && l < 24);
    const int  moff     = (l < 8) ? l : (8 + (l - 16));

    for (int tile = 0; tile < 32; ++tile) {
        const int ch = tile * 16 + m;
        float a0 = cb0, a1 = cb1;
#pragma unroll
        for (int t = 0; t < 9; ++t) {
            a0 += w0[t] * sgp[PAD + ch + (t - 4) * d0];
            a1 += w1[t] * sgp[PAD + ch + (t - 4) * d1];
        }
        a0 = fmaxf(a0, 0.f) * lw0;   // relu(conv) * lin_w  -> folded into A
        a1 = fmaxf(a1, 0.f) * lw1;
        v2f a; a.x = a0; a.y = a1;
        v8f cz = {};
        // D = A(16x4) x ones(4x16) + 0  ->  every column = sum_k feat*lin_w
        v8f d = __builtin_amdgcn_wmma_f32_16x16x4_f32(
            /*neg_a=*/false, a, /*neg_b=*/false, bones,
            /*c_mod=*/(short)0, cz, /*reuse_a=*/false, /*reuse_b=*/false);

        // select element idx (0..7) without scratch
        float v = d[0];
#pragma unroll
        for (int r = 1; r < 8; ++r) v = (idx == r) ? d[r] : v;

        if (do_store) {
            const float g = 1.0f / (1.0f + expf(-(v + lb)));
            gate[b * CTOT + tile * 16 + moff] = g;
        }
    }
}

// ---------------------------------------------------------------------------
// Kernel 3: out[b,c,:,:] = gate[b,c] * map1[b,c,:,:].
// Input loads regular-temporal (hit L2 filled by kernel 1); output stored
// non-temporal so the 134MB of stores don't evict the input from L2.
// ---------------------------------------------------------------------------
__global__ __launch_bounds__(256) void k_scale(
    const float* __restrict__ x1, const float* __restrict__ x2,
    const float* __restrict__ x3, const float* __restrict__ x4,
    const float* __restrict__ gate, float* __restrict__ out)
{
    const int plane = blockIdx.x;
    const int b = plane >> 9;
    const int c = plane & 511;
    const int sel = c >> 7;
    const float* src = (sel == 0) ? x1 : (sel == 1) ? x2 : (sel == 2) ? x3 : x4;
    src += (size_t)(b * C_PER_IN + (c & 127)) * HW;

    const float g = gate[plane];
    const v4f* ps = (const v4f*)src;
    v4f* po = (v4f*)(out + (size_t)plane * HW);
    const int t = threadIdx.x;
#pragma unroll
    for (int k = 0; k < 4; ++k) {
        v4f v = ps[t + k * 256];
        v4f r; r.x = v.x * g; r.y = v.y * g; r.z = v.z * g; r.w = v.w * g;
        __builtin_nontemporal_store(r, po + t + k * 256);
    }
}

// ---------------------------------------------------------------------------
extern "C" void kernel_launch(void* const* d_in, const int* in_sizes, int n_in,
                              void* d_out, int out_size, void* d_ws, size_t ws_size,
                              hipStream_t stream) {
    const float* x1     = (const float*)d_in[0];
    const float* x2     = (const float*)d_in[1];
    const float* x3     = (const float*)d_in[2];
    const float* x4     = (const float*)d_in[3];
    const float* conv_w = (const float*)d_in[4];
    const float* conv_b = (const float*)d_in[5];
    const float* lin_w  = (const float*)d_in[6];
    const float* lin_b  = (const float*)d_in[7];
    float* out = (float*)d_out;

    float* gp   = (float*)d_ws;        // 8192 floats
    float* gate = gp + NPLANE;         // 8192 floats

    k_reduce<<<NPLANE, 256, 0, stream>>>(x1, x2, x3, x4, gp);
    k_gate  <<<BATCH,   32, 0, stream>>>(gp, conv_w, conv_b, lin_w, lin_b, gate);
    k_scale <<<NPLANE, 256, 0, stream>>>(x1, x2, x3, x4, gate, out);
}